// GCNConvWithAdj_63574105915528
// MI455X (gfx1250) — compile-verified
//
#include <hip/hip_runtime.h>

typedef __attribute__((ext_vector_type(2))) float v2f;
typedef __attribute__((ext_vector_type(8))) float v8f;

#define GN  256   // output / B columns (fixed by problem)
#define BM  32    // block row tile
#define KC  32    // K chunk staged in LDS
#define STA 36    // padded LDS stride for A tile rows (floats)
#define STK 36    // padded LDS stride for transposed-B columns (floats)

// ---- CDNA5 async global->LDS copies (ASYNCcnt-tracked) ----
__device__ __forceinline__ unsigned lds_off(const void* p) {
  // generic pointers to LDS carry the LDS byte offset in the low 32 bits
  return (unsigned)(unsigned long long)p;
}
__device__ __forceinline__ void async_b128(unsigned lds, const void* g) {
  asm volatile("global_load_async_to_lds_b128 %0, %1, off"
               :: "v"(lds), "v"((unsigned long long)g) : "memory");
}
__device__ __forceinline__ void async_b32(unsigned lds, const void* g) {
  asm volatile("global_load_async_to_lds_b32 %0, %1, off"
               :: "v"(lds), "v"((unsigned long long)g) : "memory");
}
__device__ __forceinline__ void wait_async0() {
  asm volatile("s_wait_asynccnt 0" ::: "memory");
}

// d[row] = 1/sqrt(sum_j adj[row][j]), 0 if degree == 0.
// One wave (32 lanes) per row; 8 waves per block.
__global__ __launch_bounds__(256) void gcn_deg_kernel(const float* __restrict__ adj,
                                                      float* __restrict__ dscale,
                                                      int n) {
  const int wave = threadIdx.x >> 5;
  const int lane = threadIdx.x & 31;
  const int row  = blockIdx.x * 8 + wave;
  const float* p = adj + (size_t)row * n;
  float s = 0.f;
  for (int c = lane * 4; c < n; c += 128) {
    const float4 v = *reinterpret_cast<const float4*>(p + c);
    s += (v.x + v.y) + (v.z + v.w);
  }
#pragma unroll
  for (int off = 16; off > 0; off >>= 1) s += __shfl_xor(s, off, 32);
  if (lane == 0) dscale[row] = (s > 0.f) ? (1.0f / sqrtf(s)) : 0.f;
}

// C[m, n] = rowscale[m] * sum_k A[m,k] * B[k,n]  (+ bias[n] if bias != null)
// A: M x K row-major, B: K x 256 row-major, C: M x 256.
// Block: 256 threads = 8 waves; block tile = 32 rows x 256 cols (full N).
// Wave grid 2x4: each wave -> 16 rows x 64 cols = 4 WMMA 16x16 f32 tiles.
// Double-buffered LDS filled with async global->LDS copies; B is stored
// transposed in LDS so every WMMA fragment is a single aligned b64 LDS load.
__global__ __launch_bounds__(256) void gcn_wmma_gemm(const float* __restrict__ A,
                                                     const float* __restrict__ B,
                                                     const float* __restrict__ rowscale,
                                                     const float* __restrict__ bias,
                                                     float* __restrict__ C,
                                                     int K) {
  __shared__ float sA[2][BM * STA];   //  2 x 4.6 KB
  __shared__ float sBT[2][GN * STK];  //  2 x 36 KB  (B transposed: [n][k])

  const int tid     = threadIdx.x;
  const int lane    = tid & 31;
  const int wave    = tid >> 5;
  const int waveRow = wave >> 2;      // 0..1
  const int waveCol = wave & 3;       // 0..3
  const int half    = lane >> 4;      // 0 = lanes 0-15, 1 = lanes 16-31
  const int l16     = lane & 15;

  const int blockM = blockIdx.x * BM;
  const int r0     = waveRow * 16;    // LDS row base of this wave's stripe
  const int n0w    = waveCol * 64;    // column base within the 256-wide tile

  // A staging: 256 threads x one b128 = 32x32 floats
  const int ar = tid >> 3;            // 0..31 rows, 8 threads/row
  const int ac = (tid & 7) * 4;       // float4 column
  // B staging (transposed): wave handles rows wave*4..wave*4+3;
  // lane covers n = lane + 32*j  (global reads 128B-contiguous per wave)
  const int brB = wave * 4;

  v8f acc0 = {}, acc1 = {}, acc2 = {}, acc3 = {};

  const int nK = K / KC;

  // ---- async tile loader ----
  auto issue_tile = [&](int k0, int buf) {
    async_b128(lds_off(&sA[buf][ar * STA + ac]),
               A + (size_t)(blockM + ar) * K + k0 + ac);
#pragma unroll
    for (int r = 0; r < 4; ++r) {
      const float* grow = B + (size_t)(k0 + brB + r) * GN;
#pragma unroll
      for (int j = 0; j < 8; ++j) {
        const int n = j * 32 + lane;
        async_b32(lds_off(&sBT[buf][n * STK + brB + r]), grow + n);
      }
    }
  };

  // prologue: fill buffer 0
  issue_tile(0, 0);
  wait_async0();
  __syncthreads();

  for (int kt = 0; kt < nK; ++kt) {
    const int cur = kt & 1;
    if (kt + 1 < nK) issue_tile((kt + 1) * KC, 1 - cur);   // overlap with compute

    const float* __restrict__ la = sA[cur];
    const float* __restrict__ lb = sBT[cur];

#pragma unroll
    for (int kk = 0; kk < KC; kk += 4) {
      const int ksel = kk + half * 2;
      // A fragment 16x4: lanes 0-15 -> K=kk,kk+1 ; lanes 16-31 -> K=kk+2,kk+3
      const v2f a = *reinterpret_cast<const v2f*>(&la[(r0 + l16) * STA + ksel]);
      // B fragments 4x16 from transposed LDS: one aligned b64 each
      const v2f b0 = *reinterpret_cast<const v2f*>(&lb[(n0w +  0 + l16) * STK + ksel]);
      const v2f b1 = *reinterpret_cast<const v2f*>(&lb[(n0w + 16 + l16) * STK + ksel]);
      const v2f b2 = *reinterpret_cast<const v2f*>(&lb[(n0w + 32 + l16) * STK + ksel]);
      const v2f b3 = *reinterpret_cast<const v2f*>(&lb[(n0w + 48 + l16) * STK + ksel]);

      acc0 = __builtin_amdgcn_wmma_f32_16x16x4_f32(false, a, false, b0, (short)0, acc0, false, false);
      acc1 = __builtin_amdgcn_wmma_f32_16x16x4_f32(false, a, false, b1, (short)0, acc1, false, false);
      acc2 = __builtin_amdgcn_wmma_f32_16x16x4_f32(false, a, false, b2, (short)0, acc2, false, false);
      acc3 = __builtin_amdgcn_wmma_f32_16x16x4_f32(false, a, false, b3, (short)0, acc3, false, false);
    }

    if (kt + 1 < nK) wait_async0();   // next tile landed
    __syncthreads();                  // everyone done reading/writing buffers
  }

  // Epilogue. C/D layout: VGPR v -> row (r0 + v + half*8), col = n0 + l16.
  const int gmBase = blockM + r0 + half * 8;
  float rs[8];
#pragma unroll
  for (int v = 0; v < 8; ++v) rs[v] = rowscale[gmBase + v];

#pragma unroll
  for (int t = 0; t < 4; ++t) {
    const int gn = n0w + t * 16 + l16;
    const v8f accv = (t == 0) ? acc0 : ((t == 1) ? acc1 : ((t == 2) ? acc2 : acc3));
    const float bval = bias ? bias[gn] : 0.f;
#pragma unroll
    for (int v = 0; v < 8; ++v) {
      const int gm = gmBase + v;
      C[(size_t)gm * GN + gn] = accv[v] * rs[v] + bval;
    }
  }
}

extern "C" void kernel_launch(void* const* d_in, const int* in_sizes, int n_in,
                              void* d_out, int out_size, void* d_ws, size_t ws_size,
                              hipStream_t stream) {
  (void)in_sizes; (void)n_in; (void)out_size; (void)ws_size;
  const int N = 8192;

  const float* x    = (const float*)d_in[0];  // [8192, 256]
  const float* adj  = (const float*)d_in[1];  // [8192, 8192]
  const float* wgt  = (const float*)d_in[2];  // [256, 256]
  const float* bias = (const float*)d_in[3];  // [256]
  float* out = (float*)d_out;                 // [8192, 256]

  float* dscale  = (float*)d_ws;              // [8192]
  float* support = dscale + N;                // [8192, 256] = d[j] * (x @ W)

  // 1) d = rsqrt(rowsum(adj))           -- streams adj once (256 MB)
  gcn_deg_kernel<<<N / 8, 256, 0, stream>>>(adj, dscale, N);

  // 2) support = rowscale_d(x @ W)      -- K = 256 (tiny GEMM)
  gcn_wmma_gemm<<<N / BM, 256, 0, stream>>>(x, wgt, dscale, nullptr, support, 256);

  // 3) out = rowscale_d(adj @ support) + bias   -- K = 8192 (streams adj once)
  gcn_wmma_gemm<<<N / BM, 256, 0, stream>>>(adj, support, dscale, bias, out, N);
}